// GraphSAGE_NET_30185030156398
// MI455X (gfx1250) — compile-verified
//
#include <hip/hip_runtime.h>
#include <hip/hip_bf16.h>
#include <math.h>

// ---------------------------------------------------------------------------
// GraphSAGE (3-layer) for MI455X / gfx1250, wave32 + WMMA.
// Memory-bound: edge scatter ~5.7GB + ~717M f32 atomics dominates; GEMMs
// (~4.4 GFLOP) run on v_wmma_f32_16x16x4_f32 with LDS-staged, zero-padded,
// transposed weights (branch-free ds_load_b64 B fragments).
// ---------------------------------------------------------------------------

typedef __attribute__((ext_vector_type(8))) float v8f;
typedef __attribute__((ext_vector_type(2))) float v2f;

__device__ __forceinline__ void atomAddF(float* p, float v) {
  unsafeAtomicAdd(p, v);   // native global_atomic_add_f32 (no CAS loop)
}

// ---------------------------------------------------------------------------
// Zero a float region (grid-stride).
__global__ void zero_f32(float* __restrict__ p, long n) {
  long i = (long)blockIdx.x * blockDim.x + threadIdx.x;
  long stride = (long)gridDim.x * blockDim.x;
  for (; i < n; i += stride) p[i] = 0.0f;
}

// ---------------------------------------------------------------------------
// In-degree counts (float so the divide path is branch-free later).
__global__ void count_deg(const int* __restrict__ dst, float* __restrict__ cnt, long E) {
  long e = (long)blockIdx.x * blockDim.x + threadIdx.x;
  if (e < E) atomAddF(&cnt[dst[e]], 1.0f);
}

__global__ void inv_deg(const float* __restrict__ cnt, float* __restrict__ inv, int N) {
  int n = blockIdx.x * blockDim.x + threadIdx.x;
  if (n < N) inv[n] = 1.0f / fmaxf(cnt[n], 1.0f);
}

// ---------------------------------------------------------------------------
// Edge scatter-add: one lane per 4 features (coalesced float4 gather of the
// source row, 4 hardware f32 atomics into the destination row).
__global__ void scatter_add(const int* __restrict__ src, const int* __restrict__ dst,
                            const float* __restrict__ in, float* __restrict__ out,
                            int shift, long E) {
  long t = (long)blockIdx.x * blockDim.x + threadIdx.x;
  long total = E << shift;
  if (t >= total) return;
  long e = t >> shift;
  int  q = (int)(t & ((1 << shift) - 1));
  int  s = src[e];
  int  d = dst[e];
  const int F4 = 1 << shift;
  float4 v = ((const float4*)in)[(long)s * F4 + q];
  float* o = out + (((long)d * F4 + q) << 2);
  atomAddF(o + 0, v.x);
  atomAddF(o + 1, v.y);
  atomAddF(o + 2, v.z);
  atomAddF(o + 3, v.w);
}

// ---------------------------------------------------------------------------
// SAGE linear: out[n,:] = (agg[n,:]*inv[n]) @ Wl + x[n,:] @ Wr + bias (opt ReLU)
// Grid: N/16 blocks; block = NP/16 waves, wave w -> output cols [16w, 16w+16).
// Weights staged in LDS transposed [n][k], zero-padded to NP columns, so the
// 16x16x4 B fragment is one branch-free ds_load_b64 per MMA.
template <int K, int NOUT>
__global__ void sage_gemm(const float* __restrict__ agg, const float* __restrict__ invdeg,
                          const float* __restrict__ xin,
                          const float* __restrict__ Wl, const float* __restrict__ Wr,
                          const float* __restrict__ bias, float* __restrict__ out,
                          int doRelu) {
  constexpr int NP = (NOUT + 15) & ~15;   // padded column count
  constexpr int KP = K + 2;               // row pitch: stagger LDS banks
  __shared__ float sWl[NP * KP];
  __shared__ float sWr[NP * KP];

  for (int idx = threadIdx.x; idx < NP * K; idx += blockDim.x) {
    int n = idx / K, k = idx - n * K;
    float wl = 0.0f, wr = 0.0f;
    if (n < NOUT) { wl = Wl[k * NOUT + n]; wr = Wr[k * NOUT + n]; }
    sWl[n * KP + k] = wl;
    sWr[n * KP + k] = wr;
  }
  __syncthreads();

  const int lane = threadIdx.x & 31;
  const int wave = threadIdx.x >> 5;
  const int g    = lane >> 4;             // lane half: K {0,1} vs K {2,3}
  const int m    = lane & 15;
  const int rowb = blockIdx.x << 4;
  const int mrow = rowb + m;
  const int ncol = (wave << 4) + m;

  const float* __restrict__ pA = agg + (long)mrow * K;
  const float* __restrict__ pX = xin + (long)mrow * K;
  const float* __restrict__ bL = sWl + ncol * KP;
  const float* __restrict__ bR = sWr + ncol * KP;
  const float sA = invdeg[mrow];

  v8f c = {};
#pragma unroll 4
  for (int k = 0; k < K; k += 4) {
    const int k0 = k + 2 * g;
    v2f a, b;
    a[0] = pA[k0] * sA;  a[1] = pA[k0 + 1] * sA;
    b[0] = bL[k0];       b[1] = bL[k0 + 1];
    c = __builtin_amdgcn_wmma_f32_16x16x4_f32(false, a, false, b, (short)0, c, false, false);
  }
#pragma unroll 4
  for (int k = 0; k < K; k += 4) {
    const int k0 = k + 2 * g;
    v2f a, b;
    a[0] = pX[k0];  a[1] = pX[k0 + 1];
    b[0] = bR[k0];  b[1] = bR[k0 + 1];
    c = __builtin_amdgcn_wmma_f32_16x16x4_f32(false, a, false, b, (short)0, c, false, false);
  }

  if (ncol < NOUT) {
    const float bv = bias[ncol];
#pragma unroll
    for (int v = 0; v < 8; ++v) {
      int row = rowb + v + 8 * g;          // C/D: VGPR v -> M=v (+8 upper half)
      float val = c[v] + bv;
      if (doRelu) val = fmaxf(val, 0.0f);
      out[(long)row * NOUT + ncol] = val;
    }
  }
}

// ---------------------------------------------------------------------------
// BatchNorm stats: sums[0..C)=sum, sums[C..2C)=sumsq. Privatized accumulation.
__global__ void bn_stats(const float* __restrict__ h, float* __restrict__ sums,
                         int N, int C) {
  int rpi = blockDim.x / C;
  int c  = threadIdx.x % C;
  int ri = threadIdx.x / C;
  if (ri >= rpi) return;
  float s = 0.0f, ss = 0.0f;
  for (long r = (long)blockIdx.x * rpi + ri; r < N; r += (long)gridDim.x * rpi) {
    float v = h[r * C + c];
    s += v; ss += v * v;
  }
  atomAddF(&sums[c],     s);
  atomAddF(&sums[C + c], ss);
}

__global__ void bn_apply(float* __restrict__ h, const float* __restrict__ sums,
                         const float* __restrict__ gamma, const float* __restrict__ beta,
                         int N, int C, int doRelu) {
  long i = (long)blockIdx.x * blockDim.x + threadIdx.x;
  long total = (long)N * C;
  if (i >= total) return;
  int c = (int)(i % C);
  float mu  = sums[c] / (float)N;
  float var = sums[C + c] / (float)N - mu * mu;
  float v = (h[i] - mu) * rsqrtf(var + 1e-5f) * gamma[c] + beta[c];
  if (doRelu) v = fmaxf(v, 0.0f);
  h[i] = v;
}

// ---------------------------------------------------------------------------
// Global mean pool (sums + counts) over 20 features.
__global__ void pool_kernel(const float* __restrict__ h3, const int* __restrict__ batch,
                            float* __restrict__ pooled, float* __restrict__ gcnt, int N) {
  long t = (long)blockIdx.x * blockDim.x + threadIdx.x;
  long total = (long)N * 20;
  if (t >= total) return;
  int n = (int)(t / 20);
  int f = (int)(t % 20);
  int gidx = batch[n];
  atomAddF(&pooled[gidx * 20 + f], h3[t]);
  if (f == 0) atomAddF(&gcnt[gidx], 1.0f);
}

// 256 graphs: mean, 20->11 linear, softmax. One block of 256.
__global__ void head_kernel(const float* __restrict__ pooled, const float* __restrict__ gcnt,
                            const float* __restrict__ Wlin, const float* __restrict__ blin,
                            float* __restrict__ out, int G) {
  int gidx = blockIdx.x * blockDim.x + threadIdx.x;
  if (gidx >= G) return;
  float ic = 1.0f / fmaxf(gcnt[gidx], 1.0f);
  float p[20];
#pragma unroll
  for (int k = 0; k < 20; ++k) p[k] = pooled[gidx * 20 + k] * ic;
  float lg[11];
  float mx = -INFINITY;
#pragma unroll
  for (int cix = 0; cix < 11; ++cix) {
    float acc = blin[cix];
#pragma unroll
    for (int k = 0; k < 20; ++k) acc += p[k] * Wlin[k * 11 + cix];
    lg[cix] = acc;
    mx = fmaxf(mx, acc);
  }
  float den = 0.0f;
#pragma unroll
  for (int cix = 0; cix < 11; ++cix) { lg[cix] = __expf(lg[cix] - mx); den += lg[cix]; }
  float id = 1.0f / den;
#pragma unroll
  for (int cix = 0; cix < 11; ++cix) out[gidx * 11 + cix] = lg[cix] * id;
}

// ---------------------------------------------------------------------------
extern "C" void kernel_launch(void* const* d_in, const int* in_sizes, int n_in,
                              void* d_out, int out_size, void* d_ws, size_t ws_size,
                              hipStream_t stream) {
  const float* x    = (const float*)d_in[0];
  const int*   ei   = (const int*)  d_in[1];
  const int*   batch= (const int*)  d_in[2];
  const float* Wl1  = (const float*)d_in[3];
  const float* bl1  = (const float*)d_in[4];
  const float* Wr1  = (const float*)d_in[5];
  const float* g1   = (const float*)d_in[6];
  const float* be1  = (const float*)d_in[7];
  const float* Wl2  = (const float*)d_in[8];
  const float* bl2  = (const float*)d_in[9];
  const float* Wr2  = (const float*)d_in[10];
  const float* Wl3  = (const float*)d_in[11];
  const float* bl3  = (const float*)d_in[12];
  const float* Wr3  = (const float*)d_in[13];
  const float* g2   = (const float*)d_in[14];
  const float* be2  = (const float*)d_in[15];
  const float* Wlin = (const float*)d_in[16];
  const float* blin = (const float*)d_in[17];
  float* out = (float*)d_out;

  const long N = in_sizes[0] / 128;   // 100000 (divisible by 16)
  const long E = in_sizes[1] / 2;     // 3200000
  const int  G = 256;

  const int* src = ei;
  const int* dst = ei + E;

  // Workspace layout (floats). Accumulators first -> one zeroing pass.
  float* ws     = (float*)d_ws;
  float* agg1   = ws;                  // N*128
  float* agg2   = agg1 + N * 128;      // N*64
  float* agg3   = agg2 + N * 64;       // N*32
  float* cnt    = agg3 + N * 32;       // N
  float* gcnt   = cnt + N;             // G
  float* pooled = gcnt + G;            // G*20
  float* st1    = pooled + (long)G*20; // 128 (sum|sumsq, C=64)
  float* st2    = st1 + 128;           // 40 (C=20), padded to 64
  long   zeroN  = (st2 + 64) - agg1;
  float* invd   = st2 + 64;            // N
  float* h1     = invd + N;            // N*64
  float* h2     = h1 + N * 64;         // N*32
  float* h3     = h2 + N * 32;         // N*20

  const int TB = 256;
  zero_f32<<<2048, TB, 0, stream>>>(agg1, zeroN);
  count_deg<<<(int)((E + TB - 1) / TB), TB, 0, stream>>>(dst, cnt, E);
  inv_deg<<<(int)((N + TB - 1) / TB), TB, 0, stream>>>(cnt, invd, (int)N);

  // ---- Layer 1: SAGE 128->64, BN, ReLU
  { long tot = E << 5;
    scatter_add<<<(int)((tot + TB - 1) / TB), TB, 0, stream>>>(src, dst, x, agg1, 5, E); }
  sage_gemm<128, 64><<<(int)(N / 16), 128, 0, stream>>>(agg1, invd, x, Wl1, Wr1, bl1, h1, 0);
  bn_stats<<<256, TB, 0, stream>>>(h1, st1, (int)N, 64);
  { long tot = N * 64;
    bn_apply<<<(int)((tot + TB - 1) / TB), TB, 0, stream>>>(h1, st1, g1, be1, (int)N, 64, 1); }

  // ---- Layer 2: SAGE 64->32, ReLU
  { long tot = E << 4;
    scatter_add<<<(int)((tot + TB - 1) / TB), TB, 0, stream>>>(src, dst, h1, agg2, 4, E); }
  sage_gemm<64, 32><<<(int)(N / 16), 64, 0, stream>>>(agg2, invd, h1, Wl2, Wr2, bl2, h2, 1);

  // ---- Layer 3: SAGE 32->20, BN
  { long tot = E << 3;
    scatter_add<<<(int)((tot + TB - 1) / TB), TB, 0, stream>>>(src, dst, h2, agg3, 3, E); }
  sage_gemm<32, 20><<<(int)(N / 16), 64, 0, stream>>>(agg3, invd, h2, Wl3, Wr3, bl3, h3, 0);
  bn_stats<<<256, TB, 0, stream>>>(h3, st2, (int)N, 20);
  { long tot = N * 20;
    bn_apply<<<(int)((tot + TB - 1) / TB), TB, 0, stream>>>(h3, st2, g2, be2, (int)N, 20, 0); }

  // ---- Pool + head
  { long tot = N * 20;
    pool_kernel<<<(int)((tot + TB - 1) / TB), TB, 0, stream>>>(h3, batch, pooled, gcnt, (int)N); }
  head_kernel<<<1, TB, 0, stream>>>(pooled, gcnt, Wlin, blin, out, G);
}